// GatedCRFLoss_82832739271149
// MI455X (gfx1250) — compile-verified
//
#include <hip/hip_runtime.h>

#define SPAN 11
#define NOFF 484            // 22*22 offsets (dx!=0, dy!=0)
#define Bn   4
#define Cn   3
#define Hn   256
#define Wn   512
#define HWn  (Hn*Wn)
#define NPIX (Bn*HWn)

#define TILE 16
#define HALO (TILE + 2*SPAN)   // 38
#define SW   40                // padded LDS row stride

typedef __attribute__((ext_vector_type(2))) float v2f;
typedef __attribute__((ext_vector_type(8))) float v8f;

// 16-lane butterfly sum via ds_swizzle (SWAPX1/2/4/8); both 16-lane halves
// reduce independently, so after this every lane holds its half's total.
__device__ __forceinline__ float sum16_swz(float v) {
    v += __int_as_float(__builtin_amdgcn_ds_swizzle(__float_as_int(v), 0x041F)); // xor 1
    v += __int_as_float(__builtin_amdgcn_ds_swizzle(__float_as_int(v), 0x081F)); // xor 2
    v += __int_as_float(__builtin_amdgcn_ds_swizzle(__float_as_int(v), 0x101F)); // xor 4
    v += __int_as_float(__builtin_amdgcn_ds_swizzle(__float_as_int(v), 0x201F)); // xor 8
    return v;
}

// Wave32 dual sum-reduction through the matrix pipe.
// A = ones(16x4); B carries the 32 lane values in its first VGPR (second = 0),
// so every row of D = A*B equals the vector of B's column sums regardless of
// the exact (k,n) slot mapping. dxv[0] then holds 16 column sums per 16-lane
// half; the swizzle butterfly finishes the reduction (all lanes get the total).
__device__ __forceinline__ void wave_reduce2(float x, float y, float& sx, float& sy) {
    v2f a;  a[0] = 1.0f; a[1] = 1.0f;
    v2f bx; bx[0] = x;   bx[1] = 0.0f;
    v2f by; by[0] = y;   by[1] = 0.0f;
    v8f z = {};
    v8f dxv = __builtin_amdgcn_wmma_f32_16x16x4_f32(false, a, false, bx, (short)0, z, false, false);
    v8f dyv = __builtin_amdgcn_wmma_f32_16x16x4_f32(false, a, false, by, (short)0, z, false, false);
    sx = sum16_swz(dxv[0]);
    sy = sum16_swz(dyv[0]);
}

// ---- zero the accumulators (num[484], den[484], ce[3]) ---------------------
__global__ void k_init(float* __restrict__ acc) {
    int i = blockIdx.x * 256 + threadIdx.x;
    if (i < 2 * NOFF + 3) acc[i] = 0.0f;
}

// ---- CE: softmax + NLL + three global sums ---------------------------------
__global__ void k_ce(const float* __restrict__ logit,
                     const int*   __restrict__ target,
                     const float* __restrict__ dstmap,
                     float* __restrict__ ce_acc) {
    __shared__ float sacc[3];
    const int tid = threadIdx.x;
    if (tid < 3) sacc[tid] = 0.0f;
    __syncthreads();

    const int idx = blockIdx.x * 256 + tid;       // grid sized exactly NPIX/256
    const int b = idx / HWn;
    const int p = idx - b * HWn;
    const float* lg = logit + (size_t)b * Cn * HWn + p;
    float l0 = lg[0], l1 = lg[HWn], l2 = lg[2 * HWn];
    float mx = fmaxf(l0, fmaxf(l1, l2));
    float e0 = __expf(l0 - mx), e1 = __expf(l1 - mx), e2 = __expf(l2 - mx);
    float s  = e0 + e1 + e2;

    int t = target[idx];
    float lt = (t == 0) ? l0 : ((t == 1) ? l1 : l2);
    float lce = mx + __logf(s) - lt;              // -log_softmax[target]
    float dv  = dstmap[idx];

    float r0, r1, r2, rz;
    wave_reduce2(lce * dv, lce * (1.0f - dv), r0, r1);
    wave_reduce2(dv, 0.0f, r2, rz);
    if ((tid & 31) == 0) {
        atomicAdd(&sacc[0], r0);
        atomicAdd(&sacc[1], r1);
        atomicAdd(&sacc[2], r2);
    }
    __syncthreads();
    if (tid < 3) atomicAdd(&ce_acc[tid], sacc[tid]);
}

// ---- CRF: LDS halo tiling; each wave owns a strided subset of offsets ------
__global__ void __launch_bounds__(256)
k_crf(const float* __restrict__ logit,
      const float* __restrict__ image,
      const float* __restrict__ depth,
      const float* __restrict__ dstmap,
      float* __restrict__ gnum,
      float* __restrict__ gden) {
    // pkA = {10*r, 10*g, 10*b, 5*depth}   pkB = {y0, y1, y2, dst}
    __shared__ float4 pkA[HALO * SW];
    __shared__ float4 pkB[HALO * SW];

    const int tid  = threadIdx.x;
    const int lane = tid & 31;
    const int wv   = tid >> 5;          // 8 waves
    const int b  = blockIdx.z;
    const int y0 = blockIdx.y * TILE - SPAN;
    const int x0 = blockIdx.x * TILE - SPAN;

    const float* img = image  + (size_t)b * 3 * HWn;
    const float* lgp = logit  + (size_t)b * 3 * HWn;
    const float* dep = depth  + (size_t)b * HWn;
    const float* dm  = dstmap + (size_t)b * HWn;

    // halo load: zero-fill OOB so m = dst = 0 kills invalid pairs with finite math
    for (int cid = tid; cid < HALO * HALO; cid += 256) {
        int ly = cid / HALO, lx = cid - ly * HALO;
        int gy = y0 + ly, gx = x0 + lx;
        bool ok = (gy >= 0) & (gy < Hn) & (gx >= 0) & (gx < Wn);
        int gi = gy * Wn + gx;
        int l  = ly * SW + lx;
        float r = 0.f, g = 0.f, bl = 0.f, d0 = 0.f, ds = 0.f;
        float q0 = 0.f, q1 = 0.f, q2 = 0.f;
        if (ok) {
            r  = img[gi]; g = img[HWn + gi]; bl = img[2 * HWn + gi];
            q0 = lgp[gi]; q1 = lgp[HWn + gi]; q2 = lgp[2 * HWn + gi];
            d0 = dep[gi]; ds = dm[gi];
        }
        // inline softmax -> yhat (OOB cells give 1/3 each; dst=0 gates them out)
        float mxv = fmaxf(q0, fmaxf(q1, q2));
        float e0 = __expf(q0 - mxv), e1 = __expf(q1 - mxv), e2 = __expf(q2 - mxv);
        float inv = 1.0f / (e0 + e1 + e2);
        pkA[l] = make_float4(r * 10.0f, g * 10.0f, bl * 10.0f, d0 * 5.0f);
        pkB[l] = make_float4(e0 * inv, e1 * inv, e2 * inv, ds);
    }
    __syncthreads();

    // cache this lane's 8 chunk-center pixels in registers
    int   lcA[8];
    float cR[8], cG[8], cB[8], cD[8], cY0[8], cY1[8], cY2[8], cS1[8];
#pragma unroll
    for (int c = 0; c < 8; ++c) {
        int p  = c * 32 + lane;                       // pixel id in 16x16 tile
        int lc = ((p >> 4) + SPAN) * SW + (p & 15) + SPAN;
        lcA[c] = lc;
        float4 A = pkA[lc];
        float4 B = pkB[lc];
        cR[c] = A.x; cG[c] = A.y; cB[c] = A.z; cD[c] = A.w;
        cY0[c] = B.x; cY1[c] = B.y; cY2[c] = B.z;
        cS1[c] = B.x + B.y + B.z;
    }

    // each wave owns offsets wv, wv+8, wv+16, ... (uniform per-wave control flow)
    for (int o = wv; o < NOFF; o += 8) {
        int dxi = o / 22, dyi = o - dxi * 22;
        int dx = dxi - SPAN + (dxi >= SPAN);          // {-11..-1, 1..11}
        int dy = dyi - SPAN + (dyi >= SPAN);
        int doff  = dx * SW + dy;
        float dind = (float)(dx * dx + dy * dy) * (1.0f / 36.0f);

        float aN = 0.0f, aD = 0.0f;
#pragma unroll
        for (int c = 0; c < 8; ++c) {
            int ln = lcA[c] - doff;                   // neighbor u-d
            float4 nA = pkA[ln];
            float4 nB = pkB[ln];

            float dr = nA.x - cR[c], dg = nA.y - cG[c], db = nA.z - cB[c];
            float drgb = dr * dr + dg * dg + db * db;
            float ddp  = nA.w - cD[c];
            float kern = __expf(-0.5f * (drgb + dind)) + __expf(-0.5f * (ddp * ddp));

            float s1n = nB.x + nB.y + nB.z;
            float dot = cY0[c] * nB.x + cY1[c] * nB.y + cY2[c] * nB.z;
            float compat = cS1[c] * s1n - dot;

            float mv = nB.w;
            aN += kern * mv * compat;
            aD += mv;
        }

        float wN, wD;
        wave_reduce2(aN, aD, wN, wD);
        if (lane == 0) {
            atomicAdd(&gnum[o], wN);
            atomicAdd(&gden[o], wD);
        }
    }
}

// ---- finalize: energies -> grid mean; assemble ce_loss ---------------------
__global__ void k_finalize(const float* __restrict__ gnum,
                           const float* __restrict__ gden,
                           const float* __restrict__ ce_acc,
                           float* __restrict__ out) {
    __shared__ float red[512];
    int tid = threadIdx.x;
    float e = 0.0f;
    if (tid < NOFF) e = gnum[tid] / gden[tid];
    red[tid] = e;
    __syncthreads();
    for (int s = 256; s > 0; s >>= 1) {
        if (tid < s) red[tid] += red[tid + s];
        __syncthreads();
    }
    if (tid == 0) {
        const float N = (float)NPIX;
        float l1 = ce_acc[0] / N, l2 = ce_acc[1] / N, cnt = ce_acc[2] / N;
        out[0] = l1 * (1.0f - cnt) + l2 * cnt;   // ce_loss
        out[1] = red[0] * (1.0f / 529.0f);       // L_gcrf (23x23 grid mean)
    }
}

extern "C" void kernel_launch(void* const* d_in, const int* in_sizes, int n_in,
                              void* d_out, int out_size, void* d_ws, size_t ws_size,
                              hipStream_t stream) {
    const float* logit  = (const float*)d_in[0];
    const int*   target = (const int*)  d_in[1];
    const float* image  = (const float*)d_in[2];
    const float* depth  = (const float*)d_in[3];
    const float* dstmap = (const float*)d_in[4];
    // d_in[5] = source_map: unused by the reference.

    float* wsf = (float*)d_ws;
    float* gnum   = wsf;
    float* gden   = wsf + NOFF;
    float* ce_acc = wsf + 2 * NOFF;
    float* out    = (float*)d_out;

    k_init<<<4, 256, 0, stream>>>(wsf);
    k_ce<<<NPIX / 256, 256, 0, stream>>>(logit, target, dstmap, ce_acc);
    dim3 grid(Wn / TILE, Hn / TILE, Bn);   // 32 x 16 x 4
    k_crf<<<grid, 256, 0, stream>>>(logit, image, depth, dstmap, gnum, gden);
    k_finalize<<<1, 512, 0, stream>>>(gnum, gden, ce_acc, out);
}